// Learnable2DInterpPosEmbDivided_fixed_12137577578512
// MI455X (gfx1250) — compile-verified
//
#include <hip/hip_runtime.h>

typedef __attribute__((ext_vector_type(2))) float v2f;
typedef __attribute__((ext_vector_type(8))) float v8f;

#define DIM 1280
#define DT  (DIM / 16)   // 80 channel tiles

// workspace layout (float offsets)
#define WM32_OFF 0
#define WM16_OFF 2048
#define WM48_OFF 3072
#define TMP_OFF  6144    // tmp: up to 64 * 48 * 1280 floats

#if __has_builtin(__builtin_amdgcn_wmma_f32_16x16x4_f32)
#define HAVE_WMMA4 1
#else
#define HAVE_WMMA4 0
#endif

// jax _fill_keys_cubic_kernel (a = -0.5)
__device__ __forceinline__ float keys_cubic(float x) {
  float out = ((1.5f * x - 2.5f) * x) * x + 1.0f;
  if (x >= 1.0f) out = ((-0.5f * x + 2.5f) * x - 4.0f) * x + 2.0f;
  if (x >= 2.0f) out = 0.0f;
  return out;
}

// Build the three resize weight matrices W_s [n,64] (n = 32,16,48), matching
// jax.image.resize(method='bicubic', antialias=True): kernel_scale = 64/n,
// sample_f = (o+0.5)*64/n - 0.5, weights normalized per output position.
__global__ void build_wmats_kernel(float* __restrict__ ws) {
  int tid = blockIdx.x * blockDim.x + threadIdx.x;
  if (tid >= 96) return;
  int n, o, off;
  if (tid < 32)      { n = 32; o = tid;      off = WM32_OFF; }
  else if (tid < 48) { n = 16; o = tid - 32; off = WM16_OFF; }
  else               { n = 48; o = tid - 48; off = WM48_OFF; }
  float inv_scale = 64.0f / (float)n;       // > 1 (downsampling)
  float kscale    = inv_scale;              // antialias => max(inv_scale, 1)
  float sample_f  = ((float)o + 0.5f) * inv_scale - 0.5f;
  float w[64];
  float s = 0.0f;
  for (int i = 0; i < 64; ++i) {
    float xd = fabsf(sample_f - (float)i) / kscale;
    float v = keys_cubic(xd);
    w[i] = v;
    s += v;
  }
  float inv = (fabsf(s) > 1000.0f * 1.1920929e-7f) ? (1.0f / s) : 0.0f;
  float* row = ws + off + o * 64;
  for (int i = 0; i < 64; ++i) row[i] = w[i] * inv;
}

// Sample 3: (t=1, 64x64) -> out = x + weight, plain fp32 stream, float4.
__global__ void add_direct_kernel(const float* __restrict__ xin,
                                  const float* __restrict__ wgt,
                                  float* __restrict__ out) {
  size_t i = ((size_t)blockIdx.x * blockDim.x + threadIdx.x) * 4;
  const size_t total = (size_t)4096 * DIM;
  if (i >= total) return;
  const size_t base = (size_t)21504 * DIM;
  float4 a = *(const float4*)(xin + base + i);
  float4 w = *(const float4*)(wgt + i);
  float4 r;
  r.x = a.x + w.x; r.y = a.y + w.y; r.z = a.z + w.z; r.w = a.w + w.w;
  *(float4*)(out + base + i) = r;
}

#if HAVE_WMMA4

// Stage A: for each input row i (64): tmp[i] = W_s[n,64] x weight[i][64,1280].
// One wave per 16x16 output tile; K=64 in 16 steps of V_WMMA_F32_16X16X4_F32.
__global__ void stageA_wmma_kernel(const float* __restrict__ wgt,
                                   const float* __restrict__ wmat,
                                   float* __restrict__ tmp, int n) {
  int wave = (int)((blockIdx.x * blockDim.x + threadIdx.x) >> 5);
  int lane = threadIdx.x & 31;
  int xt = n >> 4;
  int total = 64 * xt * DT;
  if (wave >= total) return;                 // wave-uniform: EXEC stays all-ones
  int dt  = wave % DT;
  int xti = (wave / DT) % xt;
  int i   = wave / (DT * xt);
  int lo   = lane & 15;
  int koff = (lane >> 4) << 1;               // K offset 0 / 2 for lane halves
  const float* Arow = wmat + (size_t)(xti * 16 + lo) * 64 + koff;
  const float* Bcol = wgt + (size_t)i * 64 * DIM + (size_t)koff * DIM + (dt * 16 + lo);
  v8f c = {};
  #pragma unroll
  for (int k = 0; k < 64; k += 4) {
    v2f a; a.x = Arow[k];                 a.y = Arow[k + 1];
    v2f b; b.x = Bcol[(size_t)k * DIM];   b.y = Bcol[(size_t)(k + 1) * DIM];
    c = __builtin_amdgcn_wmma_f32_16x16x4_f32(false, a, false, b, (short)0, c,
                                              false, false);
  }
  int mo = (lane >> 4) << 3;                 // row offset 0 / 8 for lane halves
  float* T = tmp + (size_t)i * n * DIM + (size_t)(xti * 16) * DIM + (dt * 16 + lo);
  #pragma unroll
  for (int r = 0; r < 8; ++r) T[(size_t)(r + mo) * DIM] = c[r];
}

// Stage B (+fused epilogue): pe2d = W_s[n,64] x tmp[64, n*1280]; then for each
// frame t: out = x + pe2d + time_weight[t]. One wave per 16x16 pe2d tile.
__global__ void stageB_wmma_kernel(const float* __restrict__ xin,
                                   const float* __restrict__ tmp,
                                   const float* __restrict__ wmat,
                                   const float* __restrict__ timew,
                                   float* __restrict__ out,
                                   int n, int tcount, long base_row) {
  int wave = (int)((blockIdx.x * blockDim.x + threadIdx.x) >> 5);
  int lane = threadIdx.x & 31;
  int yt = n >> 4;
  int ctiles = n * DT;                       // tiles over composite (x,d) axis
  int total = yt * ctiles;
  if (wave >= total) return;
  int ct  = wave % ctiles;
  int yti = wave / ctiles;
  int c0   = ct * 16;
  int xcol = c0 / DIM;                       // 16 | DIM so tile stays in one x
  int d0   = c0 % DIM;
  int lo   = lane & 15;
  int koff = (lane >> 4) << 1;
  const float* Arow = wmat + (size_t)(yti * 16 + lo) * 64 + koff;
  size_t bstride = (size_t)n * DIM;
  const float* Bcol = tmp + (size_t)koff * bstride + (size_t)c0 + lo;
  v8f c = {};
  #pragma unroll
  for (int k = 0; k < 64; k += 4) {
    v2f a; a.x = Arow[k];                     a.y = Arow[k + 1];
    v2f b; b.x = Bcol[(size_t)k * bstride];   b.y = Bcol[(size_t)(k + 1) * bstride];
    c = __builtin_amdgcn_wmma_f32_16x16x4_f32(false, a, false, b, (short)0, c,
                                              false, false);
  }
  int mo   = (lane >> 4) << 3;
  int cold = d0 + lo;
  const float* xb = xin + (size_t)base_row * DIM;
  float* ob = out + (size_t)base_row * DIM;
  for (int t = 0; t < tcount; ++t) {
    float twv = timew[(size_t)t * DIM + cold];
    size_t tb = (size_t)t * n * n;
    #pragma unroll
    for (int r = 0; r < 8; ++r) {
      int y = yti * 16 + r + mo;
      size_t idx = (tb + (size_t)y * n + xcol) * DIM + cold;
      ob[idx] = xb[idx] + c[r] + twv;
    }
  }
}

#else  // fallback: scalar versions (kept only so the build never hard-fails)

__global__ void stageA_scalar_kernel(const float* __restrict__ wgt,
                                     const float* __restrict__ wmat,
                                     float* __restrict__ tmp, int n) {
  size_t e = (size_t)blockIdx.x * blockDim.x + threadIdx.x;
  size_t total = (size_t)64 * n * DIM;
  if (e >= total) return;
  int d = (int)(e % DIM);
  int x = (int)((e / DIM) % n);
  int i = (int)(e / ((size_t)DIM * n));
  const float* W = wmat + (size_t)x * 64;
  const float* B = wgt + (size_t)i * 64 * DIM + d;
  float acc = 0.0f;
  for (int j = 0; j < 64; ++j) acc = fmaf(W[j], B[(size_t)j * DIM], acc);
  tmp[e] = acc;
}

__global__ void stageB_scalar_kernel(const float* __restrict__ xin,
                                     const float* __restrict__ tmp,
                                     const float* __restrict__ wmat,
                                     const float* __restrict__ timew,
                                     float* __restrict__ out,
                                     int n, int tcount, long base_row) {
  size_t e = (size_t)blockIdx.x * blockDim.x + threadIdx.x;
  size_t total = (size_t)n * n * DIM;
  if (e >= total) return;
  int d = (int)(e % DIM);
  int x = (int)((e / DIM) % n);
  int y = (int)(e / ((size_t)DIM * n));
  const float* W = wmat + (size_t)y * 64;
  const float* B = tmp + (size_t)x * DIM + d;
  size_t bstride = (size_t)n * DIM;
  float pe = 0.0f;
  for (int i = 0; i < 64; ++i) pe = fmaf(W[i], B[(size_t)i * bstride], pe);
  const float* xb = xin + (size_t)base_row * DIM;
  float* ob = out + (size_t)base_row * DIM;
  for (int t = 0; t < tcount; ++t) {
    size_t idx = ((size_t)t * n * n + (size_t)y * n + x) * DIM + d;
    ob[idx] = xb[idx] + pe + timew[(size_t)t * DIM + d];
  }
}

#endif

extern "C" void kernel_launch(void* const* d_in, const int* in_sizes, int n_in,
                              void* d_out, int out_size, void* d_ws, size_t ws_size,
                              hipStream_t stream) {
  const float* xin  = (const float*)d_in[0];  // [25600, 1280] f32
  const float* wgt  = (const float*)d_in[1];  // [64, 64, 1280] f32
  const float* twp  = (const float*)d_in[2];  // [16, 1280] f32
  (void)d_in; (void)in_sizes; (void)n_in; (void)out_size; (void)ws_size;
  float* out = (float*)d_out;
  float* ws  = (float*)d_ws;
  float* tmp = ws + TMP_OFF;

  build_wmats_kernel<<<1, 128, 0, stream>>>(ws);

  // Sample 3: (1, 64, 64) full-res, t==1 -> out = x + weight
  add_direct_kernel<<<5120, 256, 0, stream>>>(xin, wgt, out);

  struct Samp { int n; int t; long base; int wmoff; };
  const Samp S[3] = { {32, 8, 0, WM32_OFF},
                      {16, 16, 8192, WM16_OFF},
                      {48, 4, 12288, WM48_OFF} };

  for (int s = 0; s < 3; ++s) {
    int n = S[s].n;
#if HAVE_WMMA4
    int tilesA  = 64 * (n / 16) * DT;
    int blocksA = (tilesA * 32 + 255) / 256;
    stageA_wmma_kernel<<<blocksA, 256, 0, stream>>>(wgt, ws + S[s].wmoff, tmp, n);
    int tilesB  = (n / 16) * n * DT;
    int blocksB = (tilesB * 32 + 255) / 256;
    stageB_wmma_kernel<<<blocksB, 256, 0, stream>>>(xin, tmp, ws + S[s].wmoff,
                                                    twp, out, n, S[s].t, S[s].base);
#else
    size_t eA = (size_t)64 * n * DIM;
    stageA_scalar_kernel<<<(unsigned)((eA + 255) / 256), 256, 0, stream>>>(
        wgt, ws + S[s].wmoff, tmp, n);
    size_t eB = (size_t)n * n * DIM;
    stageB_scalar_kernel<<<(unsigned)((eB + 255) / 256), 256, 0, stream>>>(
        xin, tmp, ws + S[s].wmoff, twp, out, n, S[s].t, S[s].base);
#endif
  }
}